// CustomLoss_62079457296845
// MI455X (gfx1250) — compile-verified
//
#include <hip/hip_runtime.h>

// ---------------------------------------------------------------------------
// CDNA5 / gfx1250 streaming loss kernel.
// HBM-bound (0.3 FLOP/byte, ~654 MB @ 23.3 TB/s => ~28 us floor).
//  * A and c tiles streamed via GLOBAL_LOAD_ASYNC_TO_LDS_B64 (ASYNCcnt),
//    double buffered per wave.
//  * Per-16-row dot products computed as diag(A x V^T) with chained
//    V_WMMA_F32_16X16X4_F32 (K padded 18->20); matrix pipe is otherwise idle.
//  * Deterministic two-stage reduction (no atomics).
// ---------------------------------------------------------------------------

typedef int   v2i __attribute__((ext_vector_type(2)));   // async b64 payload
typedef float v2f __attribute__((ext_vector_type(2)));   // WMMA A/B (16x4 f32)
typedef float v8f __attribute__((ext_vector_type(8)));   // WMMA C/D (16x16 f32)

__device__ __forceinline__ void async_copy_b64(const void* g, void* l) {
#if __has_builtin(__builtin_amdgcn_global_load_async_to_lds_b64)
  __builtin_amdgcn_global_load_async_to_lds_b64((v2i*)g, (v2i*)l, 0, 0);
#else
  asm volatile("global_load_async_to_lds_b64 %0, %1, off"
               :: "v"((unsigned)(size_t)l),
                  "v"((unsigned long long)(size_t)g)
               : "memory");
#endif
}

__device__ __forceinline__ void wait_async_le18() {
#if __has_builtin(__builtin_amdgcn_s_wait_asynccnt)
  __builtin_amdgcn_s_wait_asynccnt(18);
#else
  asm volatile("s_wait_asynccnt 0x12" ::: "memory");
#endif
  asm volatile("" ::: "memory");   // compiler fence: LDS reads stay below wait
}

__device__ __forceinline__ void wait_async_0() {
#if __has_builtin(__builtin_amdgcn_s_wait_asynccnt)
  __builtin_amdgcn_s_wait_asynccnt(0);
#else
  asm volatile("s_wait_asynccnt 0x0" ::: "memory");
#endif
  asm volatile("" ::: "memory");
}

#define NB_MAX      2048
#define BLK         128     // 4 waves (wave32)
#define WPB         4
#define DCOLS       18
#define GROUP_ROWS  32      // rows per wave per iteration (2 WMMA tiles of 16)
#define GRP_FLOATS  (GROUP_ROWS * DCOLS)   // 576 floats per array per group

__global__ void __launch_bounds__(BLK)
loss_stage1(const float* __restrict__ cMat, const float* __restrict__ yv,
            const float* __restrict__ Amat, const float* __restrict__ m4uc,
            const float* __restrict__ m4c,  const float* __restrict__ beta,
            const float* __restrict__ ymeanp, const float* __restrict__ yscalep,
            float* __restrict__ partials, int nGroups, int itersPerWave)
{
  // [buffer][wave][ A:0..575 | C:576..1151 ]  -> 36 KB
  __shared__ float lds[2][WPB][2 * GRP_FLOATS];
  __shared__ float wsum1[WPB], wsum2[WPB];

  const int tid  = threadIdx.x;
  const int wave = tid >> 5;
  const int lane = tid & 31;
  const long long waveGlobal = (long long)blockIdx.x * WPB + wave;

  const float ymean = ymeanp[0];
  const float inv   = 1.0f / yscalep[0];

  // ---- per-lane WMMA operand geometry (fixed for the whole kernel) ----
  // A 16x4 f32 layout: lane l -> M = l%16, VGPR v -> K = k0 + 2*(l>>4) + v.
  // B 4x16 mirrors it with N = l%16, so B[k,n] = beta[k] + c[n,k] uses the
  // SAME LDS indices as A (transpose is free via LDS indexing).
  const int mrow = lane & 15;
  const int kh   = (lane >> 4) << 1;           // 0 or 2
  float bp0[5], bp1[5];                        // per-lane beta pairs per chunk
#pragma unroll
  for (int j = 0; j < 5; ++j) {
    const int ka = j * 4 + kh;
    const int kb = ka + 1;
    bp0[j] = (ka < DCOLS) ? beta[ka] : 0.0f;
    bp1[j] = (kb < DCOLS) ? beta[kb] : 0.0f;
  }
  // D diag: D[r,r] at (VGPR r, lane r); D[r+8,r+8] at (VGPR r, lane r+24).
  const bool active = (lane < 8) || (lane >= 24);
  const int  ridx   = (lane < 8) ? lane : ((lane >= 24) ? (lane - 16) : 0);

  long long gBase = waveGlobal * itersPerWave;
  long long gEnd  = gBase + itersPerWave;
  if (gEnd > nGroups) gEnd = nGroups;

  float acc1 = 0.0f, acc2 = 0.0f;

  auto issue = [&](long long g, int buf) {
    // async LDS writes are unordered vs pending ds_loads from this wave:
    // make sure the buffer we are about to overwrite has been fully read.
    asm volatile("s_wait_dscnt 0x0" ::: "memory");
    const float* gA = Amat + g * GRP_FLOATS;
    const float* gC = cMat + g * GRP_FLOATS;
    float* dA = &lds[buf][wave][0];
    float* dC = &lds[buf][wave][GRP_FLOATS];
#pragma unroll
    for (int k = 0; k < 9; ++k) {               // 9 * 32 lanes * 2 floats = 576
      const int idx = (k * 32 + lane) * 2;
      async_copy_b64(gA + idx, dA + idx);
      async_copy_b64(gC + idx, dC + idx);
    }                                            // 18 async ops / wave / group
  };

  int buf = 0;
  if (gBase < gEnd) issue(gBase, 0);
  for (long long g = gBase; g < gEnd; ++g) {
    const bool hasNext = (g + 1) < gEnd;
    if (hasNext) issue(g + 1, buf ^ 1);          // prefetch next group
    if (hasNext) wait_async_le18();              // oldest 18 (current buf) done
    else         wait_async_0();

#pragma unroll
    for (int b = 0; b < 2; ++b) {                // two 16-row WMMA tiles
      const float* tA = &lds[buf][wave][b * 16 * DCOLS];
      const float* tC = &lds[buf][wave][GRP_FLOATS + b * 16 * DCOLS];

      v8f d = {0.f, 0.f, 0.f, 0.f, 0.f, 0.f, 0.f, 0.f};
#pragma unroll
      for (int j = 0; j < 5; ++j) {              // K = 20 (18 + 2 zero pad)
        const int ka = j * 4 + kh;
        const int kb = ka + 1;
        const int ia = mrow * DCOLS + ((ka < DCOLS) ? ka : 0);
        const int ib = mrow * DCOLS + ((kb < DCOLS) ? kb : 0);
        const float a0 = (ka < DCOLS) ? tA[ia] : 0.0f;   // zero pad => product 0
        const float a1 = (kb < DCOLS) ? tA[ib] : 0.0f;
        const float b0 = (ka < DCOLS) ? (bp0[j] + tC[ia]) : 0.0f;
        const float b1 = (kb < DCOLS) ? (bp1[j] + tC[ib]) : 0.0f;
        v2f va = {a0, a1};
        v2f vb = {b0, b1};
        d = __builtin_amdgcn_wmma_f32_16x16x4_f32(false, va, false, vb,
                                                  (short)0, d, false, false);
      }

      // extract diag -> lanes {0..7} rows +0..7, lanes {24..31} rows +8..15
      float dot = 0.0f;
#pragma unroll
      for (int r = 0; r < 8; ++r) {
        const bool pick = (lane == r) || (lane == r + 24);
        dot = pick ? d[r] : dot;
      }

      const long long row = g * GROUP_ROWS + b * 16 + ridx;
      const float yh = (m4uc[row] - dot - ymean) * inv;
      const float e1 = yh - yv[row];
      const float e2 = yh - m4c[row];
      if (active) {
        acc1 = fmaf(e1, e1, acc1);
        acc2 = fmaf(e2, e2, acc2);
      }
    }
    buf ^= 1;
  }

  // wave32 reduction
  for (int off = 16; off > 0; off >>= 1) {
    acc1 += __shfl_down(acc1, off, 32);
    acc2 += __shfl_down(acc2, off, 32);
  }
  if (lane == 0) { wsum1[wave] = acc1; wsum2[wave] = acc2; }
  __syncthreads();
  if (tid == 0) {
    float s1 = wsum1[0] + wsum1[1] + wsum1[2] + wsum1[3];
    float s2 = wsum2[0] + wsum2[1] + wsum2[2] + wsum2[3];
    partials[2 * blockIdx.x]     = s1;
    partials[2 * blockIdx.x + 1] = s2;
  }
}

__global__ void __launch_bounds__(256)
loss_stage2(const float* __restrict__ partials, int nPairs, float invN,
            float* __restrict__ out)
{
  __shared__ float sh[256];
  const int tid = threadIdx.x;
  float s = 0.0f;
  for (int i = tid; i < nPairs; i += 256)
    s += partials[2 * i] + partials[2 * i + 1];
  sh[tid] = s;
  __syncthreads();
  for (int w = 128; w > 0; w >>= 1) {
    if (tid < w) sh[tid] += sh[tid + w];
    __syncthreads();
  }
  if (tid == 0) out[0] = sh[0] * invN;   // mean1 + mean2 = (S1+S2)/N
}

extern "C" void kernel_launch(void* const* d_in, const int* in_sizes, int n_in,
                              void* d_out, int out_size, void* d_ws, size_t ws_size,
                              hipStream_t stream) {
  const float* c      = (const float*)d_in[0];
  const float* y      = (const float*)d_in[1];
  const float* A      = (const float*)d_in[2];
  const float* m4uc   = (const float*)d_in[3];
  const float* m4c    = (const float*)d_in[4];
  const float* beta   = (const float*)d_in[5];
  const float* ymean  = (const float*)d_in[6];
  const float* yscale = (const float*)d_in[7];

  const int n = in_sizes[1];               // N (y is N x 1)
  int nGroups = n / GROUP_ROWS;            // N = 4194304 is divisible by 32

  int NB = NB_MAX;
  const size_t need = (size_t)NB * 2 * sizeof(float);
  if (ws_size < need) {
    NB = (int)(ws_size / (2 * sizeof(float)));
    if (NB < 1) NB = 1;
  }
  const long long totalWaves = (long long)NB * WPB;
  int itersPerWave = (int)((nGroups + totalWaves - 1) / totalWaves);
  if (itersPerWave < 1) itersPerWave = 1;

  float* partials = (float*)d_ws;
  loss_stage1<<<NB, BLK, 0, stream>>>(c, y, A, m4uc, m4c, beta, ymean, yscale,
                                      partials, nGroups, itersPerWave);
  const float invN = (float)(1.0 / (double)n);
  loss_stage2<<<1, 256, 0, stream>>>(partials, NB, invN, (float*)d_out);
}